// CNN_19688130085198
// MI455X (gfx1250) — compile-verified
//
#include <hip/hip_runtime.h>

// ---------------------------------------------------------------------------
// SpiralConv stack on MI455X (gfx1250, wave32).
// 4x gather-GEMM layers via V_WMMA_F32_16X16X4_F32 (fp32 matches reference
// numerics; L3 = [400k,576]x[576,128] = 59 GFLOP dominates -> compute
// bound, so everything feeds the XDL pipe).  Software-pipelined: gather
// loads for spiral step s+1 overlap the WMMAs of step s via double-buffered
// LDS, one workgroup barrier per step.
// ---------------------------------------------------------------------------

typedef float v2f __attribute__((ext_vector_type(2)));
typedef float v8f __attribute__((ext_vector_type(8)));

constexpr int S_LEN = 9;        // spiral length
constexpr int BATCH = 8;
constexpr int NVERT = 50000;    // divisible by 16
constexpr int TPB   = NVERT / 16;       // 3125 row-tiles per batch
constexpr int ROWS  = BATCH * NVERT;    // 400000

// ---------------------------------------------------------------------------
// One spiral-conv layer:
//   out[b,n,:] = sum_s  x[b, idx[n,s], :] @ W[s*CIN:(s+1)*CIN, :]  + bias
// Block = 256 threads = 8 waves, handles TPBK row-tiles of 16 rows.
// Wave w: column slice colB = (w % COLT)*16, accumulates MT row-tiles so each
// B-fragment global load is reused MT times.
// ---------------------------------------------------------------------------
template<int CIN, int COUT, int KPAD, int MT, int TPBK>
__global__ __launch_bounds__(256)
void spiral_wmma(const float* __restrict__ x,      // [BATCH, NVERT, CIN]
                 const int*   __restrict__ idx,    // [NVERT, S_LEN]
                 const float* __restrict__ W,      // [S_LEN*CIN, COUT]
                 const float* __restrict__ bias,   // [COUT]
                 float* __restrict__ out)          // [BATCH, NVERT, COUT]
{
    constexpr int COLT = COUT / 16;
    static_assert((8 / COLT) * MT == TPBK, "wave tiling must cover the block");
    constexpr int STRIDE = KPAD + 4;   // +4 floats: 16B aligned, kills bank conflicts
    constexpr bool VEC = (CIN & 3) == 0;
    constexpr int KV = KPAD / 4;
    constexpr int NITEM = VEC ? (TPBK * 16 * KV) / 256 : (TPBK * 16 * KPAD) / 256;
    static_assert(VEC ? ((TPBK * 16 * KV) % 256 == 0)
                      : ((TPBK * 16 * KPAD) % 256 == 0), "exact staging split");

    __shared__ float ldsA[2][TPBK][16][STRIDE];   // double-buffered gathered A
    __shared__ int   ldsRow[TPBK][16][S_LEN];     // gather row index table
    __shared__ int   tileB[TPBK];                 // batch per tile
    __shared__ int   tileN0[TPBK];                // first vertex per tile

    const int tid   = threadIdx.x;
    const int lane  = tid & 31;
    const int wave  = tid >> 5;
    const int tile0 = blockIdx.x * TPBK;

    // ---- per-block tile coordinates: batch and first vertex per tile
    if (tid < TPBK) {
        const int gt = tile0 + tid;
        const int b  = gt / TPB;
        tileB[tid]  = b;
        tileN0[tid] = (gt - b * TPB) * 16;
    }
    __syncthreads();

    // ---- stage ALL gather row indices once (contiguous idx reads)
#pragma unroll 1
    for (int i = tid; i < TPBK * 16 * S_LEN; i += 256) {
        const int s = i % S_LEN;
        const int m = (i / S_LEN) & 15;
        const int t = i / (S_LEN * 16);
        ldsRow[t][m][s] = idx[(tileN0[t] + m) * S_LEN + s];
    }
    __syncthreads();

    const int m16   = lane & 15;            // A row / B col within the 16-tile
    const int kHi   = (lane >> 4) << 1;     // K offset of this half-wave: 0 or 2
    const int colB  = (wave % COLT) * 16;   // output-column base of this wave
    const int tBase = (wave / COLT) * MT;   // first local row-tile of this wave

    float4 vreg[VEC ? NITEM : 1];
    float  sreg[VEC ? 1 : NITEM];

    // issue gather loads for spiral step s into registers (no LDS touch)
    auto issue = [&](int s) {
        if constexpr (VEC) {
#pragma unroll
            for (int j = 0; j < NITEM; ++j) {
                const int i  = tid + j * 256;
                const int kv = i % KV;
                const int m  = (i / KV) & 15;
                const int t  = i / (KV * 16);
                const int r  = ldsRow[t][m][s];
                vreg[j] = *(const float4*)
                    (x + ((size_t)tileB[t] * NVERT + r) * CIN + kv * 4);
            }
        } else {
#pragma unroll
            for (int j = 0; j < NITEM; ++j) {
                const int i = tid + j * 256;
                const int k = i % KPAD;
                const int m = (i / KPAD) & 15;
                const int t = i / (KPAD * 16);
                float v = 0.f;
                if (k < CIN) {
                    const int r = ldsRow[t][m][s];
                    v = x[((size_t)tileB[t] * NVERT + r) * CIN + k];
                }
                sreg[j] = v;
            }
        }
    };

    // commit staged registers into LDS buffer `buf`
    auto commit = [&](int buf) {
        if constexpr (VEC) {
#pragma unroll
            for (int j = 0; j < NITEM; ++j) {
                const int i  = tid + j * 256;
                const int kv = i % KV;
                const int m  = (i / KV) & 15;
                const int t  = i / (KV * 16);
                *(float4*)(&ldsA[buf][t][m][kv * 4]) = vreg[j];
            }
        } else {
#pragma unroll
            for (int j = 0; j < NITEM; ++j) {
                const int i = tid + j * 256;
                const int k = i % KPAD;
                const int m = (i / KPAD) & 15;
                const int t = i / (KPAD * 16);
                ldsA[buf][t][m][k] = sreg[j];
            }
        }
    };

    const v8f vzero = {0.f, 0.f, 0.f, 0.f, 0.f, 0.f, 0.f, 0.f};
    v8f acc[MT];
#pragma unroll
    for (int t = 0; t < MT; ++t) acc[t] = vzero;

    issue(0);
#pragma unroll 1
    for (int s = 0; s < S_LEN; ++s) {
        const int buf = s & 1;
        commit(buf);                       // staged regs into LDS buffer buf
        if (s + 1 < S_LEN) issue(s + 1);   // prefetch next step into regs
        __syncthreads();                   // one barrier per spiral step

        // ---- K-loop: V_WMMA_F32_16X16X4_F32; MT row-tiles share each B frag
        const float* Ws = W + (size_t)(s * CIN) * COUT;
#pragma unroll
        for (int k0 = 0; k0 < KPAD; k0 += 4) {
            const int kA = k0 + kHi;
            v2f bfrag;
            bfrag.x = (kA     < CIN) ? Ws[(size_t)(kA    ) * COUT + colB + m16] : 0.f;
            bfrag.y = (kA + 1 < CIN) ? Ws[(size_t)(kA + 1) * COUT + colB + m16] : 0.f;
#pragma unroll
            for (int t = 0; t < MT; ++t) {
                v2f afrag;
                afrag.x = ldsA[buf][tBase + t][m16][kA];
                afrag.y = ldsA[buf][tBase + t][m16][kA + 1];
                acc[t] = __builtin_amdgcn_wmma_f32_16x16x4_f32(
                    /*neg_a=*/false, afrag, /*neg_b=*/false, bfrag,
                    /*c_mod=*/(short)0, acc[t],
                    /*reuse_a=*/false, /*reuse_b=*/false);
            }
        }
    }

    // ---- epilogue: bias + store (C/D layout: lane n = l&15, m = r + 8*(l>>4))
    const float bv = bias[colB + m16];
#pragma unroll
    for (int t = 0; t < MT; ++t) {
        const int lt = tBase + t;
        const int b  = tileB[lt];
        const int n0 = tileN0[lt];
#pragma unroll
        for (int r = 0; r < 8; ++r) {
            const int m = r + ((lane >> 4) << 3);
            out[(((size_t)b * NVERT) + n0 + m) * COUT + colB + m16] = acc[t][r] + bv;
        }
    }
}

// ---------------------------------------------------------------------------
// Final linear [*,128] @ [128,3] + bias : memory-bound, plain VALU.
// ---------------------------------------------------------------------------
__global__ __launch_bounds__(256)
void final_linear(const float* __restrict__ x,    // [ROWS, 128]
                  const float* __restrict__ Wf,   // [128, 3]
                  const float* __restrict__ bf,   // [3]
                  float* __restrict__ out)        // [ROWS, 3]
{
    __shared__ float w[128 * 3];
    for (int i = threadIdx.x; i < 128 * 3; i += 256) w[i] = Wf[i];
    __syncthreads();

    const int row = blockIdx.x * 256 + threadIdx.x;
    if (row >= ROWS) return;

    const float4* xr = (const float4*)(x + (size_t)row * 128);
    float a0 = bf[0], a1 = bf[1], a2 = bf[2];
#pragma unroll 8
    for (int c4 = 0; c4 < 32; ++c4) {
        const float4 v = xr[c4];
        const int c = c4 * 4;
        a0 += v.x * w[(c + 0) * 3 + 0] + v.y * w[(c + 1) * 3 + 0]
            + v.z * w[(c + 2) * 3 + 0] + v.w * w[(c + 3) * 3 + 0];
        a1 += v.x * w[(c + 0) * 3 + 1] + v.y * w[(c + 1) * 3 + 1]
            + v.z * w[(c + 2) * 3 + 1] + v.w * w[(c + 3) * 3 + 1];
        a2 += v.x * w[(c + 0) * 3 + 2] + v.y * w[(c + 1) * 3 + 2]
            + v.z * w[(c + 2) * 3 + 2] + v.w * w[(c + 3) * 3 + 2];
    }
    out[(size_t)row * 3 + 0] = a0;
    out[(size_t)row * 3 + 1] = a1;
    out[(size_t)row * 3 + 2] = a2;
}

// ---------------------------------------------------------------------------
// Launch: inputs in setup_inputs() order:
// 0:x 1:spiral_idx 2:W0 3:b0 4:W1 5:b1 6:W2 7:b2 8:W3 9:b3 10:Wf 11:bf
// ---------------------------------------------------------------------------
extern "C" void kernel_launch(void* const* d_in, const int* in_sizes, int n_in,
                              void* d_out, int out_size, void* d_ws, size_t ws_size,
                              hipStream_t stream) {
    (void)in_sizes; (void)n_in; (void)out_size; (void)ws_size;

    const float* x   = (const float*)d_in[0];
    const int*   idx = (const int*)  d_in[1];
    const float* W0  = (const float*)d_in[2];
    const float* b0  = (const float*)d_in[3];
    const float* W1  = (const float*)d_in[4];
    const float* b1  = (const float*)d_in[5];
    const float* W2  = (const float*)d_in[6];
    const float* b2  = (const float*)d_in[7];
    const float* W3  = (const float*)d_in[8];
    const float* b3  = (const float*)d_in[9];
    const float* Wf  = (const float*)d_in[10];
    const float* bf  = (const float*)d_in[11];

    // intermediate activations in workspace
    float* x1 = (float*)d_ws;                       // [ROWS,16]   25.6 MB
    float* x2 = x1 + (size_t)ROWS * 16;             // [ROWS,32]   51.2 MB
    float* x3 = x2 + (size_t)ROWS * 32;             // [ROWS,64]  102.4 MB
    float* x4 = x3 + (size_t)ROWS * 64;             // [ROWS,128] 204.8 MB

    dim3 blk(256);
    const int tiles = BATCH * TPB;                  // 25000 row-tiles

    spiral_wmma< 3,  16,  4, 1, 8><<<tiles / 8, blk, 0, stream>>>(x,  idx, W0, b0, x1);
    spiral_wmma<16,  32, 16, 2, 8><<<tiles / 8, blk, 0, stream>>>(x1, idx, W1, b1, x2);
    spiral_wmma<32,  64, 32, 4, 8><<<tiles / 8, blk, 0, stream>>>(x2, idx, W2, b2, x3);
    spiral_wmma<64, 128, 64, 4, 4><<<tiles / 4, blk, 0, stream>>>(x3, idx, W3, b3, x4);

    final_linear<<<(ROWS + 255) / 256, blk, 0, stream>>>(x4, Wf, bf, (float*)d_out);
}